// GNN_VN_Hierarchical_89094801588812
// MI455X (gfx1250) — compile-verified
//
#include <hip/hip_runtime.h>
#include <hip/hip_bf16.h>
#include <hip/hip_fp16.h>

#define NNODES 100000
#define NEDGES 1600000
#define HDIM   64
#define HNUMB  128
#define NCONVL 2
#define NMLPL  2
#define NEGSLP 0.2f

typedef __attribute__((ext_vector_type(2))) float v2f;
typedef __attribute__((ext_vector_type(8))) float v8f;

__device__ __forceinline__ float leaky(float x) { return x > 0.f ? x : NEGSLP * x; }

// monotone float <-> uint encoding for atomicMax on floats (handles negatives)
__device__ __forceinline__ unsigned fenc(float f) {
    unsigned u = __float_as_uint(f);
    return (u & 0x80000000u) ? ~u : (u | 0x80000000u);
}
__device__ __forceinline__ float fdec(unsigned u) {
    return (u & 0x80000000u) ? __uint_as_float(u & 0x7fffffffu) : __uint_as_float(~u);
}

// ---------------------------------------------------------------------------
// GEMM: Hout[N,64] = X[N,64] @ W[64,64] via V_WMMA_F32_16X16X4_F32.
// One wave = 16-row strip, block = 8 waves = 128 rows.
// W staged into LDS pair-swizzled: Wlds[(k>>1)*128 + n*2 + (k&1)] = W[k][n],
// so each lane's B fragment (2 consecutive K values, one column) is one
// aligned 8-byte ds_load_b64 straight into an even VGPR pair.
// ---------------------------------------------------------------------------
__global__ __launch_bounds__(256) void gemm_xw_kernel(const float* __restrict__ X,
                                                      const float* __restrict__ W,
                                                      float* __restrict__ Hout,
                                                      int nrows) {
    __shared__ float Wlds[HDIM * HDIM];   // 16 KB, swizzled
    const int tid = threadIdx.x;
    for (int idx = tid; idx < HDIM * HDIM; idx += 256) {
        int k = idx >> 6, n = idx & 63;                     // coalesced global read
        Wlds[(k >> 1) * 128 + n * 2 + (k & 1)] = W[idx];    // swizzled LDS write
    }
    __syncthreads();

    const int wave    = tid >> 5;
    const int lane    = tid & 31;
    const int hi      = lane >> 4;   // which half of the wave
    const int mrow    = lane & 15;
    const int rowbase = blockIdx.x * 128 + wave * 16;
    if (rowbase >= nrows) return;    // wave-uniform guard: EXEC stays all-ones

    // A fragments (f32 16x4 per k-step): lanes 0-15 hold K=4k+0,1; lanes 16-31 K=4k+2,3
    v2f afrag[16];
#pragma unroll
    for (int kk = 0; kk < 16; ++kk) {
        const v2f* ap = (const v2f*)(X + (size_t)(rowbase + mrow) * HDIM + kk * 4 + hi * 2);
        afrag[kk] = *ap;
    }
#pragma unroll
    for (int nt = 0; nt < 4; ++nt) {
        const int ncol = nt * 16 + mrow;
        v8f acc = {};
#pragma unroll
        for (int kk = 0; kk < 16; ++kk) {
            // B fragment (4x16): lane (hi,mrow) needs K = 4k+2*hi+{0,1}, N = ncol.
            // Swizzled pair index = (4k+2*hi)>>1 = 2k+hi  -> one contiguous b64.
            v2f b = *(const v2f*)&Wlds[(kk * 2 + hi) * 128 + ncol * 2];
            acc = __builtin_amdgcn_wmma_f32_16x16x4_f32(false, afrag[kk], false, b,
                                                        (short)0, acc, false, false);
        }
#pragma unroll
        for (int v = 0; v < 8; ++v)
            Hout[(size_t)(rowbase + v + 8 * hi) * HDIM + ncol] = acc[v];
    }
}

// a_s[r] = h[r,:]·a_src ; a_d[r] = h[r,:]·a_dst
__global__ void dots_kernel(const float* __restrict__ Hm, const float* __restrict__ asrc,
                            const float* __restrict__ adst, float* __restrict__ a_s,
                            float* __restrict__ a_d, int n) {
    int r = blockIdx.x * blockDim.x + threadIdx.x;
    if (r >= n) return;
    const float* hr = Hm + (size_t)r * HDIM;
    float s = 0.f, d = 0.f;
#pragma unroll
    for (int k = 0; k < HDIM; ++k) { float v = hr[k]; s += v * asrc[k]; d += v * adst[k]; }
    a_s[r] = s; a_d[r] = d;
}

// self-loop: e_self = leaky(a_s+a_d); seed segment max
__global__ void node_init_kernel(const float* __restrict__ a_s, const float* __restrict__ a_d,
                                 float* __restrict__ e_self, unsigned* __restrict__ m_enc, int n) {
    int i = blockIdx.x * blockDim.x + threadIdx.x;
    if (i >= n) return;
    float es = leaky(a_s[i] + a_d[i]);
    e_self[i] = es;
    m_enc[i]  = fenc(es);
}

__global__ void edge_max_kernel(const int* __restrict__ src, const int* __restrict__ dst,
                                const float* __restrict__ a_s, const float* __restrict__ a_d,
                                unsigned* __restrict__ m_enc, int ne) {
    int e = blockIdx.x * blockDim.x + threadIdx.x;
    if (e >= ne) return;
    int s = src[e], d = dst[e];
    atomicMax(m_enc + d, fenc(leaky(a_s[s] + a_d[d])));
}

// denom starts as the self-loop weight
__global__ void node_selfw_kernel(const float* __restrict__ e_self,
                                  const unsigned* __restrict__ m_enc,
                                  float* __restrict__ denom, int n) {
    int i = blockIdx.x * blockDim.x + threadIdx.x;
    if (i >= n) return;
    denom[i] = __expf(e_self[i] - fdec(m_enc[i]));
}

// num[n,f] = w_self[n] * h[n,f]
__global__ void num_init_kernel(const float* __restrict__ wself, const float* __restrict__ Hm,
                                float* __restrict__ num, int total) {
    int t = blockIdx.x * blockDim.x + threadIdx.x;
    if (t >= total) return;
    num[t] = wself[t >> 6] * Hm[t];
}

__global__ void edge_w_kernel(const int* __restrict__ src, const int* __restrict__ dst,
                              const float* __restrict__ a_s, const float* __restrict__ a_d,
                              const unsigned* __restrict__ m_enc, float* __restrict__ w_edge,
                              float* __restrict__ denom, int ne) {
    int e = blockIdx.x * blockDim.x + threadIdx.x;
    if (e >= ne) return;
    int s = src[e], d = dst[e];
    float w = __expf(leaky(a_s[s] + a_d[d]) - fdec(m_enc[d]));
    w_edge[e] = w;
    atomicAdd(denom + d, w);
}

// one thread per (edge, feature): coalesced h gather + contiguous atomic scatter
__global__ void edge_scatter_kernel(const int* __restrict__ src, const int* __restrict__ dst,
                                    const float* __restrict__ w_edge, const float* __restrict__ Hm,
                                    float* __restrict__ num, long total) {
    long t = (long)blockIdx.x * blockDim.x + threadIdx.x;
    if (t >= total) return;
    int e = (int)(t >> 6), f = (int)(t & 63);
    atomicAdd(num + (size_t)dst[e] * HDIM + f, w_edge[e] * Hm[(size_t)src[e] * HDIM + f]);
}

__global__ void finalize_kernel(const float* __restrict__ num, const float* __restrict__ denom,
                                const float* __restrict__ bias, float* __restrict__ out, int total) {
    int t = blockIdx.x * blockDim.x + threadIdx.x;
    if (t >= total) return;
    out[t] = num[t] / (denom[t >> 6] + 1e-16f) + bias[t & 63];
}

__global__ void gather_add_kernel(const float* __restrict__ out, const float* __restrict__ vnd,
                                  const int* __restrict__ blocks, float* __restrict__ x_in, int total) {
    int t = blockIdx.x * blockDim.x + threadIdx.x;
    if (t >= total) return;
    x_in[t] = out[t] + vnd[blocks[t >> 6] * HDIM + (t & 63)];
}

__global__ void zero_kernel(float* __restrict__ p, int n) {
    int t = blockIdx.x * blockDim.x + threadIdx.x;
    if (t < n) p[t] = 0.f;
}

// segment-sum over 128 blocks: LDS-accumulated, then flushed with global atomics
__global__ __launch_bounds__(256) void seg_sum_kernel(const float* __restrict__ out,
                                                      const int* __restrict__ blocks,
                                                      float* __restrict__ vn_seg, long total) {
    __shared__ float acc[HNUMB * HDIM];   // 32 KB
    for (int i = threadIdx.x; i < HNUMB * HDIM; i += 256) acc[i] = 0.f;
    __syncthreads();
    for (long t = (long)blockIdx.x * blockDim.x + threadIdx.x; t < total;
         t += (long)gridDim.x * blockDim.x) {
        int n = (int)(t >> 6), f = (int)(t & 63);
        atomicAdd(&acc[blocks[n] * HDIM + f], out[t]);
    }
    __syncthreads();
    for (int i = threadIdx.x; i < HNUMB * HDIM; i += 256) atomicAdd(vn_seg + i, acc[i]);
}

__global__ __launch_bounds__(256) void vn_update_kernel(float* __restrict__ vn_direct,
                                                        const float* __restrict__ vn_seg,
                                                        float* __restrict__ vn_root) {
    int tid = threadIdx.x;
    for (int i = tid; i < HNUMB * HDIM; i += 256)
        vn_direct[i] += vn_seg[i] + vn_root[i & (HDIM - 1)];
    __syncthreads();
    if (tid < HDIM) {
        float s = 0.f;
        for (int r = 0; r < HNUMB; ++r) s += vn_direct[r * HDIM + tid];
        vn_root[tid] += s;
    }
}

// two-layer MLP with ReLU, one row per block (in-place safe: row read before write)
__global__ __launch_bounds__(64) void mlp_kernel(float* __restrict__ vn,
                                                 const float* __restrict__ W1, const float* __restrict__ b1,
                                                 const float* __restrict__ W2, const float* __restrict__ b2) {
    __shared__ float rin[HDIM];
    __shared__ float t1[HDIM];
    int f = threadIdx.x;
    float* row = vn + (size_t)blockIdx.x * HDIM;
    rin[f] = row[f];
    __syncthreads();
    float s = b1[f];
#pragma unroll
    for (int k = 0; k < HDIM; ++k) s += rin[k] * W1[k * HDIM + f];
    t1[f] = fmaxf(s, 0.f);
    __syncthreads();
    float s2 = b2[f];
#pragma unroll
    for (int k = 0; k < HDIM; ++k) s2 += t1[k] * W2[k * HDIM + f];
    row[f] = fmaxf(s2, 0.f);
}

__global__ void vn_init_kernel(const float* __restrict__ vn_emb, float* __restrict__ vn_direct,
                               float* __restrict__ vn_root) {
    int t = blockIdx.x * blockDim.x + threadIdx.x;
    if (t < HNUMB * HDIM) vn_direct[t] = vn_emb[t & (HDIM - 1)];
    if (t < HDIM) vn_root[t] = vn_emb[t];
}

// ---------------------------------------------------------------------------
extern "C" void kernel_launch(void* const* d_in, const int* in_sizes, int n_in,
                              void* d_out, int out_size, void* d_ws, size_t ws_size,
                              hipStream_t stream) {
    const float* x       = (const float*)d_in[0];
    const int*   ei      = (const int*)d_in[1];
    const int*   src     = ei;
    const int*   dst     = ei + NEDGES;
    const int*   hblocks = (const int*)d_in[2];
    // d_in[3] = h_levels (unused), d_in[4] = h_num (compile-time 128)
    const float* W0      = (const float*)d_in[5];
    const float* asrc0   = (const float*)d_in[6];
    const float* adst0   = (const float*)d_in[7];
    const float* b0      = (const float*)d_in[8];
    const float* Wl      = (const float*)d_in[9];
    const float* asrcl   = (const float*)d_in[10];
    const float* adstl   = (const float*)d_in[11];
    const float* bl      = (const float*)d_in[12];
    const float* vn_emb  = (const float*)d_in[13];
    const float* mW1     = (const float*)d_in[14];
    const float* mb1     = (const float*)d_in[15];
    const float* mW2     = (const float*)d_in[16];
    const float* mb2     = (const float*)d_in[17];
    float* out = (float*)d_out;

    // workspace carve (all 256B aligned)
    char*  ws  = (char*)d_ws;
    size_t off = 0;
    auto carve = [&](size_t bytes) -> void* {
        void* p = ws + off;
        off += (bytes + 255) & ~(size_t)255;
        return p;
    };
    float*    h         = (float*)carve((size_t)NNODES * HDIM * 4);
    float*    x_in      = (float*)carve((size_t)NNODES * HDIM * 4);
    float*    num       = (float*)carve((size_t)NNODES * HDIM * 4);
    float*    a_s       = (float*)carve((size_t)NNODES * 4);
    float*    a_d       = (float*)carve((size_t)NNODES * 4);
    float*    e_self    = (float*)carve((size_t)NNODES * 4);
    float*    denom     = (float*)carve((size_t)NNODES * 4);
    unsigned* m_enc     = (unsigned*)carve((size_t)NNODES * 4);
    float*    w_edge    = (float*)carve((size_t)NEDGES * 4);
    float*    vn_direct = (float*)carve((size_t)HNUMB * HDIM * 4);
    float*    vn_root   = (float*)carve((size_t)HDIM * 4);
    float*    vn_seg    = (float*)carve((size_t)HNUMB * HDIM * 4);

    const int NF      = NNODES * HDIM;           // 6.4M
    const int nodeBlk = (NNODES + 255) / 256;
    const int nfBlk   = NF / 256;                // exact
    const int edgeBlk = NEDGES / 256;            // exact
    const long EF     = (long)NEDGES * HDIM;     // 102.4M
    const int efBlk   = (int)(EF / 256);         // exact

    auto run_gat = [&](const float* Xin, const float* W, const float* asrc,
                       const float* adst, const float* bias) {
        gemm_xw_kernel<<<(NNODES + 127) / 128, 256, 0, stream>>>(Xin, W, h, NNODES);
        dots_kernel<<<nodeBlk, 256, 0, stream>>>(h, asrc, adst, a_s, a_d, NNODES);
        node_init_kernel<<<nodeBlk, 256, 0, stream>>>(a_s, a_d, e_self, m_enc, NNODES);
        edge_max_kernel<<<edgeBlk, 256, 0, stream>>>(src, dst, a_s, a_d, m_enc, NEDGES);
        node_selfw_kernel<<<nodeBlk, 256, 0, stream>>>(e_self, m_enc, denom, NNODES);
        num_init_kernel<<<nfBlk, 256, 0, stream>>>(denom, h, num, NF);
        edge_w_kernel<<<edgeBlk, 256, 0, stream>>>(src, dst, a_s, a_d, m_enc, w_edge, denom, NEDGES);
        edge_scatter_kernel<<<efBlk, 256, 0, stream>>>(src, dst, w_edge, h, num, EF);
        finalize_kernel<<<nfBlk, 256, 0, stream>>>(num, denom, bias, out, NF);
    };

    vn_init_kernel<<<(HNUMB * HDIM + 255) / 256, 256, 0, stream>>>(vn_emb, vn_direct, vn_root);

    // initial GATConv(input -> hidden)
    run_gat(x, W0, asrc0, adst0, b0);

    for (int l = 0; l < NCONVL; ++l) {
        gather_add_kernel<<<nfBlk, 256, 0, stream>>>(out, vn_direct, hblocks, x_in, NF);
        run_gat(x_in, Wl + l * HDIM * HDIM, asrcl + l * HDIM, adstl + l * HDIM, bl + l * HDIM);
        zero_kernel<<<(HNUMB * HDIM + 255) / 256, 256, 0, stream>>>(vn_seg, HNUMB * HDIM);
        seg_sum_kernel<<<128, 256, 0, stream>>>(out, hblocks, vn_seg, (long)NF);
        vn_update_kernel<<<1, 256, 0, stream>>>(vn_direct, vn_seg, vn_root);
        for (int j = 0; j < NMLPL; ++j) {
            mlp_kernel<<<HNUMB, 64, 0, stream>>>(vn_direct, mW1 + j * HDIM * HDIM, mb1 + j * HDIM,
                                                 mW2 + j * HDIM * HDIM, mb2 + j * HDIM);
            mlp_kernel<<<1, 64, 0, stream>>>(vn_root, mW1 + j * HDIM * HDIM, mb1 + j * HDIM,
                                             mW2 + j * HDIM * HDIM, mb2 + j * HDIM);
        }
    }
}